// P_A_GIN_79517024518359
// MI455X (gfx1250) — compile-verified
//
#include <hip/hip_runtime.h>

#define N_NODES 4096
#define A_ANCH  144
#define F_DIM   128
#define O_DIM   128

typedef float v2f __attribute__((ext_vector_type(2)));
typedef float v8f __attribute__((ext_vector_type(8)));

// ---------------------------------------------------------------------------
// Kernel 1: P = feature @ lh_w[:, :F].T          [N, O]
//           Q = feature @ lh_w[:, F:].T + lh_b   [N, O]
// fp32 WMMA 16x16x4. One wave -> one 16x16 tile of BOTH P and Q (shared A).
// Block = 256 threads = 8 waves; wave w owns column tile w*16 (O=128 -> 8).
// Grid.x = N/16 row tiles.
// ---------------------------------------------------------------------------
__global__ __launch_bounds__(256) void precompute_gemm(
    const float* __restrict__ feature,   // [N, F]   row-major
    const float* __restrict__ lh_w,      // [O, 2F]  row-major
    const float* __restrict__ lh_b,      // [O]
    float* __restrict__ P,               // [N, O]
    float* __restrict__ Q)               // [N, O]
{
  const int lane  = threadIdx.x & 31;
  const int wave  = threadIdx.x >> 5;        // 0..7 -> column tile
  const int m     = lane & 15;               // row/col within tile
  const int khalf = (lane >> 4) << 1;        // lanes 0-15 -> K{0,1}; 16-31 -> K{2,3}
  const int row   = blockIdx.x * 16 + m;     // node index (A operand)
  const int col   = wave * 16 + m;           // output channel (B operand)

  const float* arow = feature + (size_t)row * F_DIM;
  const float* bl   = lh_w + (size_t)col * (2 * F_DIM);        // left  half: k in [0,128)
  const float* br   = bl + F_DIM;                               // right half: k in [128,256)

  v8f accP = {};
  v8f accQ = {};
  for (int k0 = 0; k0 < F_DIM; k0 += 4) {
    v2f a  = *(const v2f*)(arow + k0 + khalf);   // A[m][k0+khalf .. +1]
    v2f bL = *(const v2f*)(bl   + k0 + khalf);   // B[k][col] = lh_w[col][k]
    v2f bR = *(const v2f*)(br   + k0 + khalf);
    accP = __builtin_amdgcn_wmma_f32_16x16x4_f32(
        false, a, false, bL, (short)0, accP, false, false);
    accQ = __builtin_amdgcn_wmma_f32_16x16x4_f32(
        false, a, false, bR, (short)0, accQ, false, false);
  }

  // D layout: VGPR r -> (lanes 0-15: M=r, lanes 16-31: M=r+8), N = lane&15
  const float qb    = lh_b[col];
  const int   rbase = blockIdx.x * 16 + ((lane >> 4) << 3);
#pragma unroll
  for (int r = 0; r < 8; ++r) {
    P[(size_t)(rbase + r) * O_DIM + col] = accP[r];
    Q[(size_t)(rbase + r) * O_DIM + col] = accQ[r] + qb;
  }
}

// ---------------------------------------------------------------------------
// Kernel 2: fused per-node message pass.  One block (160 threads) per node.
//   phase 1 (thread = anchor): d[n,a] via 1->O->1 MLP (weights in LDS),
//       out_position[n,a] = lo_b + sum_o relu(d*P[g][o] + Q[n][o]) * lo_w[o]
//   phase 2 (thread = channel o): out_structure[n,o] =
//       mean_a relu(d[a]*P[g[a]][o] + Q[n][o])   (coalesced column reads)
// P rows are 2 MB total -> hot in L2; HBM traffic is negligible.
// ---------------------------------------------------------------------------
__global__ __launch_bounds__(160) void fused_message(
    const float* __restrict__ dists_max,  // [N, A]
    const int*   __restrict__ argmax,     // [N, A]
    const float* __restrict__ score,      // [A]
    const float* __restrict__ dc_w1,      // [O]
    const float* __restrict__ dc_b1,      // [O]
    const float* __restrict__ dc_w2,      // [O]
    const float* __restrict__ dc_b2,      // [1]
    const float* __restrict__ lo_w,       // [O]
    const float* __restrict__ lo_b,       // [1]
    const float* __restrict__ P,          // [N, O]
    const float* __restrict__ Q,          // [N, O]
    float* __restrict__ out_pos,          // [N, A]
    float* __restrict__ out_struct)       // [N, O]
{
  __shared__ float sq[O_DIM], slo[O_DIM], sw1[O_DIM], sb1[O_DIM], sw2[O_DIM];
  __shared__ float sd[A_ANCH];
  __shared__ int   sg[A_ANCH];

  const int n   = blockIdx.x;
  const int tid = threadIdx.x;

  if (tid < O_DIM) {
    sq[tid]  = Q[(size_t)n * O_DIM + tid];
    slo[tid] = lo_w[tid];
    sw1[tid] = dc_w1[tid];
    sb1[tid] = dc_b1[tid];
    sw2[tid] = dc_w2[tid];
  }
  __syncthreads();

  if (tid < A_ANCH) {
    // ---- scalar MLP: d = (sum_o relu(x*w1[o]+b1[o]) * w2[o] + b2) * score[a]
    const float x = dists_max[(size_t)n * A_ANCH + tid];
    float dacc = 0.f;
#pragma unroll 8
    for (int o = 0; o < O_DIM; ++o)
      dacc += fmaxf(fmaf(x, sw1[o], sb1[o]), 0.f) * sw2[o];
    const float d = (dacc + dc_b2[0]) * score[tid];
    const int   g = argmax[(size_t)n * A_ANCH + tid];
    sd[tid] = d;
    sg[tid] = g;

    // ---- out_position: row-wise dot over the gathered P row (L2-resident)
    const float4* prow = (const float4*)(P + (size_t)g * O_DIM);
    float pos = lo_b[0];
#pragma unroll 4
    for (int oc = 0; oc < O_DIM / 4; ++oc) {
      const float4 p  = prow[oc];
      const float* qv = sq  + oc * 4;
      const float* lw = slo + oc * 4;
      const float h0 = fmaxf(fmaf(d, p.x, qv[0]), 0.f);
      const float h1 = fmaxf(fmaf(d, p.y, qv[1]), 0.f);
      const float h2 = fmaxf(fmaf(d, p.z, qv[2]), 0.f);
      const float h3 = fmaxf(fmaf(d, p.w, qv[3]), 0.f);
      pos = fmaf(h0, lw[0], pos);
      pos = fmaf(h1, lw[1], pos);
      pos = fmaf(h2, lw[2], pos);
      pos = fmaf(h3, lw[3], pos);
    }
    out_pos[(size_t)n * A_ANCH + tid] = pos;
  }
  __syncthreads();

  if (tid < O_DIM) {
    // ---- out_structure: column-wise mean over anchors (coalesced in o)
    const float q = sq[tid];
    float s = 0.f;
#pragma unroll 8
    for (int a = 0; a < A_ANCH; ++a) {
      const float p = P[(size_t)sg[a] * O_DIM + tid];
      s += fmaxf(fmaf(sd[a], p, q), 0.f);
    }
    out_struct[(size_t)n * O_DIM + tid] = s * (1.0f / (float)A_ANCH);
  }
}

// ---------------------------------------------------------------------------
extern "C" void kernel_launch(void* const* d_in, const int* in_sizes, int n_in,
                              void* d_out, int out_size, void* d_ws, size_t ws_size,
                              hipStream_t stream) {
  (void)in_sizes; (void)n_in; (void)out_size; (void)ws_size;

  const float* feature      = (const float*)d_in[0];
  const float* dists_max    = (const float*)d_in[1];
  const int*   dists_argmax = (const int*)  d_in[2];
  const float* score        = (const float*)d_in[3];
  const float* dc_w1        = (const float*)d_in[4];
  const float* dc_b1        = (const float*)d_in[5];
  const float* dc_w2        = (const float*)d_in[6];
  const float* dc_b2        = (const float*)d_in[7];
  const float* lh_w         = (const float*)d_in[8];
  const float* lh_b         = (const float*)d_in[9];
  const float* lo_w         = (const float*)d_in[10];
  const float* lo_b         = (const float*)d_in[11];

  float* P = (float*)d_ws;                       // [N, O]  2 MB
  float* Q = P + (size_t)N_NODES * O_DIM;        // [N, O]  2 MB

  float* out_pos    = (float*)d_out;                         // [N, A]
  float* out_struct = out_pos + (size_t)N_NODES * A_ANCH;    // [N, O]

  precompute_gemm<<<N_NODES / 16, 256, 0, stream>>>(feature, lh_w, lh_b, P, Q);
  fused_message<<<N_NODES, 160, 0, stream>>>(
      dists_max, dists_argmax, score, dc_w1, dc_b1, dc_w2, dc_b2,
      lo_w, lo_b, P, Q, out_pos, out_struct);
}